// EncodingLayer_88699664597510
// MI455X (gfx1250) — compile-verified
//
#include <hip/hip_runtime.h>

typedef __attribute__((ext_vector_type(2))) float v2f;
typedef __attribute__((ext_vector_type(8))) float v8f;
typedef __attribute__((ext_vector_type(4))) int   v4i;

// Problem constants (from reference): B=16, H*W=1024, D=128, K=32
#define DDIM   128
#define KCODE  32
#define NPB    1024           // rows (pixels) per batch
#define CHROWS 64             // rows per chunk (one workgroup)
#define NCH    (NPB / CHROWS) // 16 chunks per batch
#define XLD    132            // LDS row stride for X   (132 % 64 == 4 -> conflict-free col reads, float4-aligned)
#define CLD    132            // LDS row stride for codewords
#define ALD    34             // LDS row stride for xc/A

// gfx1250 async global->LDS path (ASYNCcnt-tracked), with safe fallback.
#if defined(__has_builtin)
#  if __has_builtin(__builtin_amdgcn_global_load_async_to_lds_b128)
#    define HAVE_ASYNC_LDS 1
#  endif
#endif
#ifndef HAVE_ASYNC_LDS
#  define HAVE_ASYNC_LDS 0
#endif

#if HAVE_ASYNC_LDS
typedef __attribute__((address_space(1))) v4i* gv4i_p;  // global (AS1) 16B vector ptr
typedef __attribute__((address_space(3))) v4i* lv4i_p;  // LDS (AS3) 16B vector ptr
__device__ __forceinline__ void async_wait0() {
#if defined(__has_builtin) && __has_builtin(__builtin_amdgcn_s_wait_asynccnt)
    __builtin_amdgcn_s_wait_asynccnt(0);
#else
    asm volatile("s_wait_asynccnt 0" ::: "memory");
#endif
}
#endif

__global__ __launch_bounds__(128)
void enc_chunk_kernel(const float* __restrict__ x,
                      const float* __restrict__ cw,
                      const float* __restrict__ scale,
                      float* __restrict__ part,
                      float* __restrict__ out,
                      int use_atomic)
{
    __shared__ float Xs[CHROWS * XLD];   // X chunk  [64][128]
    __shared__ float Cs[KCODE * CLD];    // codewords [32][128]
    __shared__ float As[CHROWS * ALD];   // xc, then softmax A  [64][32]
    __shared__ float x2s[CHROWS];
    __shared__ float c2s[KCODE];
    __shared__ float scl[KCODE];
    __shared__ float colsum[KCODE];

    const int tid   = threadIdx.x;
    const int chunk = blockIdx.x;   // 0..NCH-1
    const int b     = blockIdx.y;   // 0..B-1
    const float* xsrc = x + ((size_t)b * NPB + (size_t)chunk * CHROWS) * DDIM;

    // ---- stage X chunk and codewords into LDS ----
#if HAVE_ASYNC_LDS
    {
        // async direct global->LDS, 16B per lane per op; no VGPR round trip
        for (int i = tid; i < CHROWS * (DDIM / 4); i += 128) {
            int r  = i >> 5;           // DDIM/4 == 32 float4 per row
            int c4 = i & 31;
            __builtin_amdgcn_global_load_async_to_lds_b128(
                (gv4i_p)(xsrc + (size_t)r * DDIM + c4 * 4),
                (lv4i_p)&Xs[r * XLD + c4 * 4], 0, 0);
        }
        for (int i = tid; i < KCODE * (DDIM / 4); i += 128) {
            int r  = i >> 5;
            int c4 = i & 31;
            __builtin_amdgcn_global_load_async_to_lds_b128(
                (gv4i_p)(cw + (size_t)r * DDIM + c4 * 4),
                (lv4i_p)&Cs[r * CLD + c4 * 4], 0, 0);
        }
        if (tid < KCODE) scl[tid] = scale[tid];
        async_wait0();
    }
#else
    {
        const float4* src = (const float4*)xsrc;
        for (int i = tid; i < CHROWS * (DDIM / 4); i += 128) {
            int r  = i >> 5;
            int c4 = i & 31;
            float4 v = src[(size_t)r * 32 + c4];
            float* dst = &Xs[r * XLD + c4 * 4];
            dst[0] = v.x; dst[1] = v.y; dst[2] = v.z; dst[3] = v.w;
        }
        const float4* csrc = (const float4*)cw;
        for (int i = tid; i < KCODE * (DDIM / 4); i += 128) {
            int r  = i >> 5;
            int c4 = i & 31;
            float4 v = csrc[r * 32 + c4];
            float* dst = &Cs[r * CLD + c4 * 4];
            dst[0] = v.x; dst[1] = v.y; dst[2] = v.z; dst[3] = v.w;
        }
        if (tid < KCODE) scl[tid] = scale[tid];
    }
#endif
    __syncthreads();

    // ---- per-row |x|^2 (threads 0..63) and per-codeword |c|^2 (threads 64..95), in parallel ----
    if (tid < CHROWS) {
        const float* row = &Xs[tid * XLD];
        float s = 0.f;
        #pragma unroll 8
        for (int d = 0; d < DDIM; ++d) s += row[d] * row[d];
        x2s[tid] = s;
    } else if (tid < CHROWS + KCODE) {
        const int k = tid - CHROWS;
        const float* row = &Cs[k * CLD];
        float s = 0.f;
        #pragma unroll 8
        for (int d = 0; d < DDIM; ++d) s += row[d] * row[d];
        c2s[k] = s;
    }
    __syncthreads();

    const int wave = tid >> 5;
    const int lane = tid & 31;
    const int half = lane >> 4;     // half-wave select
    const int mn   = lane & 15;     // M (A) / N (B) index within tile
    const int koff = half * 2;      // K offset for f32 16x16x4 fragments

    // ---- GEMM1: xc[r][k] = X[r][:] . C[k][:]   (8 tiles of 16x16, 2 per wave) ----
    for (int t = 0; t < 2; ++t) {
        const int idx = wave * 2 + t;      // 0..7
        const int rt  = idx >> 1;          // row tile 0..3
        const int ct  = idx & 1;           // col tile 0..1
        v8f acc = {};
        const float* arow = &Xs[(rt * 16 + mn) * XLD];
        const float* brow = &Cs[(ct * 16 + mn) * CLD];
        #pragma unroll 8
        for (int kb = 0; kb < DDIM; kb += 4) {
            v2f a, bm;
            a[0]  = arow[kb + koff];
            a[1]  = arow[kb + koff + 1];
            bm[0] = brow[kb + koff];       // B[k][n] = C[n][k]
            bm[1] = brow[kb + koff + 1];
            acc = __builtin_amdgcn_wmma_f32_16x16x4_f32(
                false, a, false, bm, (short)0, acc, false, false);
        }
        #pragma unroll
        for (int r = 0; r < 8; ++r)
            As[(rt * 16 + half * 8 + r) * ALD + ct * 16 + mn] = acc[r];
    }
    __syncthreads();

    // ---- softmax over K per row: A = softmax(scale_k * (x2 - 2*xc + c2_k)) ----
    if (tid < CHROWS) {
        float slv[KCODE];
        float* arow = &As[tid * ALD];
        const float xx = x2s[tid];
        float mx = -3.402823466e38f;
        #pragma unroll
        for (int k = 0; k < KCODE; ++k) {
            float v = scl[k] * (xx - 2.f * arow[k] + c2s[k]);
            slv[k] = v;
            mx = fmaxf(mx, v);
        }
        float sum = 0.f;
        #pragma unroll
        for (int k = 0; k < KCODE; ++k) {
            float e = __expf(slv[k] - mx);
            slv[k] = e;
            sum += e;
        }
        const float inv = 1.f / sum;
        #pragma unroll
        for (int k = 0; k < KCODE; ++k) arow[k] = slv[k] * inv;
    }
    __syncthreads();

    // ---- chunk-partial column sums of A ----
    if (tid < KCODE) {
        float s = 0.f;
        #pragma unroll 8
        for (int n = 0; n < CHROWS; ++n) s += As[n * ALD + tid];
        colsum[tid] = s;
    }
    __syncthreads();

    // ---- GEMM2: Epart[k][d] = sum_n A[n][k] * X[n][d]  (16 tiles, 4 per wave) ----
    for (int t = 0; t < 4; ++t) {
        const int idx = wave * 4 + t;  // 0..15
        const int mt  = idx >> 3;      // codeword tile 0..1
        const int nt  = idx & 7;       // d tile 0..7
        v8f acc = {};
        #pragma unroll 8
        for (int kb = 0; kb < CHROWS; kb += 4) {
            v2f a, bm;
            a[0]  = As[(kb + koff)     * ALD + mt * 16 + mn]; // A_mat[m=k][kk=row]
            a[1]  = As[(kb + koff + 1) * ALD + mt * 16 + mn];
            bm[0] = Xs[(kb + koff)     * XLD + nt * 16 + mn]; // B[kk=row][n=d]
            bm[1] = Xs[(kb + koff + 1) * XLD + nt * 16 + mn];
            acc = __builtin_amdgcn_wmma_f32_16x16x4_f32(
                false, a, false, bm, (short)0, acc, false, false);
        }
        #pragma unroll
        for (int r = 0; r < 8; ++r) {
            const int k_ = mt * 16 + half * 8 + r;
            const int d_ = nt * 16 + mn;
            const float val = acc[r] - colsum[k_] * Cs[k_ * CLD + d_];
            if (use_atomic) {
                atomicAdd(&out[((size_t)b * KCODE + k_) * DDIM + d_], val);
            } else {
                part[(((size_t)b * NCH + chunk) * KCODE + k_) * DDIM + d_] = val;
            }
        }
    }
}

// Deterministic fixed-order reduction of chunk partials.
__global__ __launch_bounds__(256)
void enc_reduce_kernel(const float* __restrict__ part, float* __restrict__ out, int total)
{
    const int i = blockIdx.x * blockDim.x + threadIdx.x;
    if (i >= total) return;
    const int b = i >> 12;        // K*D = 4096 elements per batch
    const int r = i & 4095;
    const float* p = part + ((size_t)b * NCH) * 4096 + r;
    float s = 0.f;
    #pragma unroll
    for (int j = 0; j < NCH; ++j) s += p[(size_t)j * 4096];
    out[i] = s;
}

extern "C" void kernel_launch(void* const* d_in, const int* in_sizes, int n_in,
                              void* d_out, int out_size, void* d_ws, size_t ws_size,
                              hipStream_t stream)
{
    (void)n_in;
    const float* x  = (const float*)d_in[0];   // [B, H, W, D] f32
    const float* cw = (const float*)d_in[1];   // [K, D] f32
    const float* sc = (const float*)d_in[2];   // [K] f32
    float* out = (float*)d_out;                // [B, K, D] f32

    const int B = in_sizes[0] / (NPB * DDIM);  // 16

    const size_t need = (size_t)B * NCH * KCODE * DDIM * sizeof(float); // 4 MB for B=16
    const int use_atomic = (ws_size < need) ? 1 : 0;

    if (use_atomic) {
        (void)hipMemsetAsync(d_out, 0, (size_t)out_size * sizeof(float), stream);
    }

    dim3 grid(NCH, B);
    enc_chunk_kernel<<<grid, 128, 0, stream>>>(x, cw, sc, (float*)d_ws, out, use_atomic);

    if (!use_atomic) {
        const int total = B * KCODE * DDIM;
        enc_reduce_kernel<<<(total + 255) / 256, 256, 0, stream>>>((const float*)d_ws, out, total);
    }
}